// CenterLoss_79448305041527
// MI455X (gfx1250) — compile-verified
//
#include <hip/hip_runtime.h>
#include <hip/hip_bf16.h>

// CenterLoss, MI455X (gfx1250, wave32).
// loss = (1/B) * sum_i clamp(||x_i - c_{label_i}||^2, 1e-12, 1e12) + (C-1)*1e-12
// Only the gathered diagonal is computed; the full B x C distmat of the
// reference is analytically folded away. Dot products run on the native f32
// tensor path (V_WMMA_F32_16X16X4_F32) to match reference f32 precision.

typedef __attribute__((ext_vector_type(2))) float v2f;
typedef __attribute__((ext_vector_type(8))) float v8f;

#define CL_B 1024
#define CL_C 100000
#define CL_D 256

__global__ void center_loss_init(float* __restrict__ out) {
    // Analytic clamp-floor contribution from the (C-1) masked-out columns
    // per row: B*(C-1)*1e-12 / B = (C-1)*1e-12.
    out[0] = (float)((double)(CL_C - 1) * 1e-12);
}

__global__ __launch_bounds__(32) void center_loss_wmma(
    const float* __restrict__ x,        // [B, D]
    const int*   __restrict__ labels,   // [B]
    const float* __restrict__ centers,  // [C, D]
    float* __restrict__ out) {
    const int lane  = threadIdx.x;       // 0..31 (wave32)
    const int tile  = blockIdx.x;        // 64 tiles of 16 rows
    const int row16 = lane & 15;         // matrix row this lane feeds
    const int half  = lane >> 4;         // K-pair selector per A/B layout
    const int r     = tile * 16 + row16;

    const int lbl = labels[r];
    // A fragment source: x[r][k0 + 2*half + {0,1}]
    const float* __restrict__ xr = x + (size_t)r * CL_D + 2 * half;
    // B fragment source: gathered center row, same K striping
    const float* __restrict__ cr = centers + (size_t)lbl * CL_D + 2 * half;

    v8f acc = {0.f, 0.f, 0.f, 0.f, 0.f, 0.f, 0.f, 0.f};
    float sx = 0.f;   // partial ||x_r||^2 (this lane's half of the K range)
    float sc = 0.f;   // partial ||c_{label_r}||^2

    // K loop: 256 / 4 = 64 WMMA_F32_16X16X4_F32 accumulations.
    // Uniform control flow: EXEC is all-ones at every WMMA.
#pragma unroll 8
    for (int k0 = 0; k0 < CL_D; k0 += 4) {
        const float2 av = *(const float2*)(xr + k0);
        const float2 bv = *(const float2*)(cr + k0);
        v2f a; a.x = av.x; a.y = av.y;
        v2f b; b.x = bv.x; b.y = bv.y;
        acc = __builtin_amdgcn_wmma_f32_16x16x4_f32(
            /*neg_a=*/false, a, /*neg_b=*/false, b,
            /*c_mod=*/(short)0, acc, /*reuse_a=*/false, /*reuse_b=*/false);
        sx = __builtin_fmaf(av.x, av.x, sx);
        sx = __builtin_fmaf(av.y, av.y, sx);
        sc = __builtin_fmaf(bv.x, bv.x, sc);
        sc = __builtin_fmaf(bv.y, bv.y, sc);
    }

    // Complete the norms: lane L holds K = 0,1 mod 4 (L<16) or 2,3 mod 4.
    const float sxf = sx + __shfl_xor(sx, 16, 32);
    const float scf = sc + __shfl_xor(sc, 16, 32);

    // Diagonal extraction from the 16x16 f32 accumulator layout:
    //   lanes 0-7  : diag m = lane,      component m
    //   lanes 24-31: diag m = lane - 16, component m - 8
    const int c7 = lane & 7;
    float dot;
    switch (c7) {
        case 0: dot = acc[0]; break;
        case 1: dot = acc[1]; break;
        case 2: dot = acc[2]; break;
        case 3: dot = acc[3]; break;
        case 4: dot = acc[4]; break;
        case 5: dot = acc[5]; break;
        case 6: dot = acc[6]; break;
        default: dot = acc[7]; break;
    }

    const bool valid = (lane < 8) || (lane >= 24);
    float val = 0.f;
    if (valid) {
        // ||x||^2 + ||c||^2 - 2 x.c, clamped exactly as the reference does
        // for the label column, pre-divided by B.
        float d = sxf + scf - 2.0f * dot;
        d = fminf(fmaxf(d, 1e-12f), 1e12f);
        val = d * (1.0f / (float)CL_B);
    }

    // Wave32 sum reduction; lane 0 commits the tile's contribution.
#pragma unroll
    for (int off = 16; off > 0; off >>= 1)
        val += __shfl_xor(val, off, 32);
    if (lane == 0)
        atomicAdd(out, val);
}

extern "C" void kernel_launch(void* const* d_in, const int* in_sizes, int n_in,
                              void* d_out, int out_size, void* d_ws, size_t ws_size,
                              hipStream_t stream) {
    (void)in_sizes; (void)n_in; (void)out_size; (void)d_ws; (void)ws_size;
    const float* x       = (const float*)d_in[0];
    const int*   labels  = (const int*)d_in[1];
    const float* centers = (const float*)d_in[2];
    float* out = (float*)d_out;

    center_loss_init<<<1, 1, 0, stream>>>(out);
    center_loss_wmma<<<CL_B / 16, 32, 0, stream>>>(x, labels, centers, out);
}